// RelationNet_78993038508755
// MI455X (gfx1250) — compile-verified
//
#include <hip/hip_runtime.h>
#include <math.h>

// ---------------------------------------------------------------------------
// RelationNet forward on gfx1250 (MI455X), fp32 path via V_WMMA_F32_16X16X4_F32
//
// Shapes: B=16, n_way=5, n_shot=n_pseudo=5, n_query=150, C=64, H=W=5
// Factorization: conv(concat(q,p)) = conv(q, W[:, :C]) + conv(p, W[:, C:])
// N-tiles overlap ([0,16) and [9,25)) -> all fragment accesses in-bounds.
// B operands stored K-pair-interleaved (float2) so every fragment is a single
// aligned b64 load straight into a contiguous VGPR pair (no shuffle moves).
// ---------------------------------------------------------------------------

typedef __attribute__((ext_vector_type(2))) float v2f;
typedef __attribute__((ext_vector_type(8))) float v8f;

#define LEAK 0.1f
#define BSTR 25           // im2col N stride (25 spatial positions)
#define NT1  9            // second N-tile origin (cols 9..24)

#if defined(__AMDGCN__) && __has_builtin(__builtin_amdgcn_global_load_async_to_lds_b128)
#define HAVE_ASYNC_LDS 1
typedef int v4i __attribute__((vector_size(16)));
typedef __attribute__((address_space(1))) v4i gas_v4i;   // global AS
typedef __attribute__((address_space(3))) v4i las_v4i;   // LDS AS
#endif

__device__ __forceinline__ v8f wmma4(v2f a, v2f b, v8f c) {
  // D = A(16x4) * B(4x16) + C(16x16), fp32
  return __builtin_amdgcn_wmma_f32_16x16x4_f32(
      /*neg_a=*/false, a, /*neg_b=*/false, b,
      /*c_mod=*/(short)0, c, /*reuse_a=*/false, /*reuse_b=*/false);
}

__device__ __forceinline__ float leaky(float x) { return x > 0.f ? x : LEAK * x; }

// im2col gather: k = ic*9 + kh*3 + kw, n = py*5 + px, 3x3 SAME conv on 5x5.
__device__ __forceinline__ float im2col_val(const float* isrc, int k, int n) {
  int ic = k / 9, r = k - ic * 9;
  int kh = r / 3, kw = r - kh * 3;
  int iy = n / 5 + kh - 1, ix = (n % 5) + kw - 1;
  return (iy >= 0 && iy < 5 && ix >= 0 && ix < 5) ? isrc[ic * 25 + iy * 5 + ix]
                                                  : 0.f;
}

// ---------------------------------------------------------------------------
// Kernel 1: protos = mean over (5 shot + 5 pseudo) supports
// ---------------------------------------------------------------------------
__global__ void protos_kernel(const float* __restrict__ x_shot,
                              const float* __restrict__ x_pseudo,
                              float* __restrict__ protos) {
  int i = blockIdx.x * blockDim.x + threadIdx.x;
  if (i >= 80 * 1600) return;
  int bw  = i / 1600;          // b*5 + way
  int rem = i - bw * 1600;     // c*25 + pos
  const float* ps = x_shot   + (size_t)(bw * 5) * 1600 + rem;
  const float* pp = x_pseudo + (size_t)(bw * 5) * 1600 + rem;
  float s = 0.f;
#pragma unroll
  for (int k = 0; k < 5; ++k) { s += ps[k * 1600]; s += pp[k * 1600]; }
  protos[i] = s * 0.1f;
}

// ---------------------------------------------------------------------------
// Kernel 1b: repack w1 [64,300] into K-pair float2: w1p[kp][col] = {w1[2kp][col],
//            w1[2kp+1][col]}  (32*300 float2)
// ---------------------------------------------------------------------------
__global__ void packw1_kernel(const float* __restrict__ w1,
                              v2f* __restrict__ w1p) {
  int i = blockIdx.x * blockDim.x + threadIdx.x;
  if (i >= 32 * 300) return;
  int kp = i / 300, col = i - kp * 300;
  v2f v;
  v.x = w1[(2 * kp) * 300 + col];
  v.y = w1[(2 * kp + 1) * 300 + col];
  w1p[i] = v;
}

// ---------------------------------------------------------------------------
// Kernel 2: per-image implicit-GEMM 3x3 SAME conv (64 in-ch half, 64 out-ch)
//   M=64 (4 wave-tiles), N=25 (two overlapping 16-wide tiles), K=576.
//   is_query==0: imgs=protos, part=1  -> raw conv maps to convP [80][64][25]
//   is_query==1: imgs=x_query, part=0 -> fuse +convP, BN, leaky, HW-mean pool
// ---------------------------------------------------------------------------
__global__ __launch_bounds__(128) void conv_fused_kernel(
    const float* __restrict__ imgs, const float* __restrict__ conv_w,
    int part, int is_query,
    const float* __restrict__ convP,
    const float* __restrict__ gamma, const float* __restrict__ beta,
    const float* __restrict__ mean,  const float* __restrict__ var,
    float* __restrict__ out) {
  __shared__ v2f   bmat[288 * BSTR];   // 57.6 KB: K-pair-interleaved im2col
  __shared__ float imgbuf[1600];       //  6.4 KB: raw image staging

  const int idx = blockIdx.x;
  const float* img = imgs + (size_t)idx * 1600;

#if defined(HAVE_ASYNC_LDS)
  // Async-copy the raw image (1600 floats) into LDS, 16B per lane per op.
  for (int e = threadIdx.x; e < 400; e += blockDim.x) {
    __builtin_amdgcn_global_load_async_to_lds_b128(
        (gas_v4i*)(unsigned long long)(img + e * 4),
        (las_v4i*)(unsigned int)(unsigned long long)(imgbuf + e * 4),
        0, 0);
  }
#if __has_builtin(__builtin_amdgcn_s_wait_asynccnt)
  __builtin_amdgcn_s_wait_asynccnt(0);
#else
  asm volatile("s_wait_asynccnt 0x0" ::: "memory");
#endif
  __syncthreads();
  const float* isrc = imgbuf;                // build im2col from LDS copy
#else
  const float* isrc = img;                   // fallback: read global directly
#endif

  // Build K-pair-interleaved im2col: bmat[kp*25+n] = {B[2kp][n], B[2kp+1][n]}
  for (int e = threadIdx.x; e < 288 * 25; e += blockDim.x) {
    int kp = e / 25, n = e - kp * 25;
    v2f v;
    v.x = im2col_val(isrc, 2 * kp, n);
    v.y = im2col_val(isrc, 2 * kp + 1, n);
    bmat[e] = v;
  }
  __syncthreads();

  const int lane = threadIdx.x & 31;
  const int wave = threadIdx.x >> 5;
  const int m    = lane & 15;      // A row (oc within tile) / B,D column
  const int hi   = lane >> 4;      // lane-half: K pair (A) / row+8 (D)
  const int oc   = wave * 16 + m;
  const float* wrow = conv_w + (size_t)oc * 1152 + part * 576;  // contiguous K
  __builtin_prefetch(wrow, 0, 0);

  v8f acc0 = {0.f, 0.f, 0.f, 0.f, 0.f, 0.f, 0.f, 0.f};
  v8f acc1 = {0.f, 0.f, 0.f, 0.f, 0.f, 0.f, 0.f, 0.f};
  for (int k0 = 0; k0 < 576; k0 += 4) {
    v2f a = *(const v2f*)(wrow + k0 + 2 * hi);   // A[m][klo], A[m][klo+1]
    int kb = ((k0 >> 1) + hi) * BSTR;            // K-pair row in bmat
    v2f b0 = bmat[kb + m];                       // tile0: col = m
    v2f b1 = bmat[kb + NT1 + m];                 // tile1: col = 9 + m (<25)
    acc0 = wmma4(a, b0, acc0);
    acc1 = wmma4(a, b1, acc1);
  }

  if (!is_query) {
    // Proto half: store raw conv output [idx][oc][25] (tile overlap benign:
    // identical WMMA inputs -> bitwise-identical values).
    float* dst = out + (size_t)idx * 1600;
#pragma unroll
    for (int r = 0; r < 8; ++r) {
      int row = wave * 16 + hi * 8 + r;          // D layout: VGPR r -> M=r/r+8
      dst[row * 25 + m]       = acc0[r];
      dst[row * 25 + NT1 + m] = acc1[r];
    }
  } else {
    // Query half: park conv map in LDS, fuse BN + leaky + pool vs 5 protos.
    __syncthreads();
    float* ylds = (float*)bmat;                  // reuse im2col buffer, 64x25
#pragma unroll
    for (int r = 0; r < 8; ++r) {
      int row = wave * 16 + hi * 8 + r;
      ylds[row * 25 + m]       = acc0[r];
      ylds[row * 25 + NT1 + m] = acc1[r];
    }
    __syncthreads();
    if (threadIdx.x < 64) {
      int c = threadIdx.x;
      int b = idx / 150;
      float g   = gamma[c];
      float inv = rsqrtf(var[c] + 1e-5f);
      float sc  = g * inv;
      float sh  = beta[c] - g * mean[c] * inv;
      for (int w = 0; w < 5; ++w) {
        const float* yp = convP + (((size_t)(b * 5 + w)) * 64 + c) * 25;
        float s = 0.f;
        for (int n = 0; n < 25; ++n)
          s += leaky(sc * (ylds[c * 25 + n] + yp[n]) + sh);
        out[((size_t)idx * 5 + w) * 64 + c] = s * 0.04f;  // mean over 25
      }
    }
  }
}

// ---------------------------------------------------------------------------
// Kernel 3: hidden = leaky(pooled @ w1 + b1), pooled [12000,64], w1p packed.
//   WMMA GEMM: 750 M-tiles x 19 N-tiles; tile 18 overlaps (starts at 284) so
//   all loads/stores stay in-bounds without guards. K=64 -> 16 WMMA steps.
// ---------------------------------------------------------------------------
__global__ __launch_bounds__(128) void mlp1_kernel(
    const float* __restrict__ pooled, const v2f* __restrict__ w1p,
    const float* __restrict__ b1, float* __restrict__ hidden) {
  int tile = blockIdx.x * 4 + (threadIdx.x >> 5);
  if (tile >= 750 * 19) return;
  int mt = tile / 19, nt = tile - mt * 19;
  int lane = threadIdx.x & 31;
  int m = lane & 15, hi = lane >> 4;
  int col = ((nt < 18) ? nt * 16 : 284) + m;     // always < 300
  const float* arow = pooled + (size_t)(mt * 16 + m) * 64;
  const v2f* brow = w1p + (size_t)hi * 300 + col;
  v8f acc = {0.f, 0.f, 0.f, 0.f, 0.f, 0.f, 0.f, 0.f};
  for (int k0 = 0; k0 < 64; k0 += 4) {
    v2f a = *(const v2f*)(arow + k0 + 2 * hi);
    v2f bf = brow[(k0 >> 1) * 300];              // w1p[(k0/2 + hi)*300 + col]
    acc = wmma4(a, bf, acc);
  }
  float bb = b1[col];
#pragma unroll
  for (int r = 0; r < 8; ++r) {
    int row = mt * 16 + hi * 8 + r;
    hidden[(size_t)row * 300 + col] = leaky(acc[r] + bb);
  }
}

// ---------------------------------------------------------------------------
// Kernel 4: out = sigmoid(hidden @ w2 + b2), hidden [12000,300], w2 [300]
// ---------------------------------------------------------------------------
__global__ void mlp2_kernel(const float* __restrict__ hidden,
                            const float* __restrict__ w2,
                            const float* __restrict__ b2,
                            float* __restrict__ out) {
  int row = blockIdx.x * blockDim.x + threadIdx.x;
  if (row >= 12000) return;
  const float* h = hidden + (size_t)row * 300;
  float s = 0.f;
  for (int j = 0; j < 300; ++j) s += h[j] * w2[j];
  s += b2[0];
  out[row] = 1.f / (1.f + expf(-s));
}

// ---------------------------------------------------------------------------
extern "C" void kernel_launch(void* const* d_in, const int* in_sizes, int n_in,
                              void* d_out, int out_size, void* d_ws, size_t ws_size,
                              hipStream_t stream) {
  (void)in_sizes; (void)n_in; (void)out_size; (void)ws_size;
  const float* x_shot   = (const float*)d_in[0];
  const float* x_pseudo = (const float*)d_in[1];
  const float* x_query  = (const float*)d_in[2];
  const float* conv_w   = (const float*)d_in[3];
  const float* gamma    = (const float*)d_in[4];
  const float* beta     = (const float*)d_in[5];
  const float* bn_mean  = (const float*)d_in[6];
  const float* bn_var   = (const float*)d_in[7];
  const float* w1       = (const float*)d_in[8];
  const float* b1       = (const float*)d_in[9];
  const float* w2       = (const float*)d_in[10];
  const float* b2       = (const float*)d_in[11];
  float* out = (float*)d_out;

  float* ws     = (float*)d_ws;
  float* protos = ws;                  // 80*1600   = 128000 floats
  float* convP  = ws + 128000;         // 80*1600   = 128000 floats
  float* pooled = ws + 256000;         // 12000*64  = 768000 floats
  float* hidden = ws + 1024000;        // 12000*300 = 3600000 floats
  v2f*   w1p    = (v2f*)(ws + 4624000);// 32*300 float2 (8B-aligned offset)

  protos_kernel<<<(128000 + 255) / 256, 256, 0, stream>>>(x_shot, x_pseudo, protos);
  packw1_kernel<<<(9600 + 255) / 256, 256, 0, stream>>>(w1, w1p);
  // Proto half of the conv (part=1), raw outputs.
  conv_fused_kernel<<<80, 128, 0, stream>>>(protos, conv_w, 1, 0,
                                            nullptr, nullptr, nullptr, nullptr,
                                            nullptr, convP);
  // Query half (part=0), fused with BN/leaky/pool over the 5 proto maps.
  conv_fused_kernel<<<2400, 128, 0, stream>>>(x_query, conv_w, 0, 1,
                                              convP, gamma, beta, bn_mean,
                                              bn_var, pooled);
  mlp1_kernel<<<(750 * 19 + 3) / 4, 128, 0, stream>>>(pooled, w1p, b1, hidden);
  mlp2_kernel<<<(12000 + 255) / 256, 256, 0, stream>>>(hidden, w2, b2, out);
}